// GCN_29386166239874
// MI455X (gfx1250) — compile-verified
//
#include <hip/hip_runtime.h>

// GCN pipeline for MI455X (gfx1250, wave32).
// Scatter/gather phases: edge-parallel f32 atomics (L2-resident working set,
// 192MB L2 >> 110MB scratch). Dense matmuls: V_WMMA_F32_16X16X4_F32 tiles,
// exact fp32, epilogues fused.

typedef __attribute__((ext_vector_type(2))) float v2f;
typedef __attribute__((ext_vector_type(8))) float v8f;

// ---------------- utility kernels ----------------

__global__ __launch_bounds__(256) void k_zero(float* __restrict__ p, size_t n) {
    size_t i = (size_t)blockIdx.x * blockDim.x + threadIdx.x;
    if (i < n) p[i] = 0.0f;
}

__global__ __launch_bounds__(256) void k_copy(float* __restrict__ o,
                                              const float* __restrict__ in, size_t n) {
    size_t i = (size_t)blockIdx.x * blockDim.x + threadIdx.x;
    if (i < n) o[i] = in[i];
}

__global__ __launch_bounds__(256) void k_degree(const int* __restrict__ src,
                                                const int* __restrict__ dst,
                                                float* __restrict__ deg_out,
                                                float* __restrict__ deg_in, int E) {
    int e = blockIdx.x * blockDim.x + threadIdx.x;
    if (e < E) {
        atomicAdd(&deg_out[src[e]], 1.0f);
        atomicAdd(&deg_in[dst[e]], 1.0f);
    }
}

__global__ __launch_bounds__(256) void k_rsqrt(float* __restrict__ p, size_t n) {
    size_t i = (size_t)blockIdx.x * blockDim.x + threadIdx.x;
    if (i < n) p[i] = rsqrtf(fmaxf(p[i], 1.0f));
}

// out[dst[e]][f] += in[src[e]][f], F=64. Two waves per edge, coalesced.
__global__ __launch_bounds__(256) void k_scatter64(const float* __restrict__ in,
                                                   const int* __restrict__ src,
                                                   const int* __restrict__ dst,
                                                   float* __restrict__ out, int E) {
    size_t t = (size_t)blockIdx.x * blockDim.x + threadIdx.x;
    int e = (int)(t >> 6);
    int f = (int)(t & 63);
    if (e < E) {
        int s = src[e], d = dst[e];
        atomicAdd(&out[(size_t)d * 64 + f], in[(size_t)s * 64 + f]);
    }
}

// GraphConv1 aggregation with the norm_s prescale folded into the gather.
__global__ __launch_bounds__(256) void k_scatter64_scaled(const float* __restrict__ in,
                                                          const float* __restrict__ norm_s,
                                                          const int* __restrict__ src,
                                                          const int* __restrict__ dst,
                                                          float* __restrict__ out, int E) {
    size_t t = (size_t)blockIdx.x * blockDim.x + threadIdx.x;
    int e = (int)(t >> 6);
    int f = (int)(t & 63);
    if (e < E) {
        int s = src[e], d = dst[e];
        atomicAdd(&out[(size_t)d * 64 + f], in[(size_t)s * 64 + f] * norm_s[s]);
    }
}

// F=32 scatter for GraphConv2 aggregation. One wave per edge.
__global__ __launch_bounds__(256) void k_scatter32(const float* __restrict__ in,
                                                   const int* __restrict__ src,
                                                   const int* __restrict__ dst,
                                                   float* __restrict__ out, int E) {
    size_t t = (size_t)blockIdx.x * blockDim.x + threadIdx.x;
    int e = (int)(t >> 5);
    int f = (int)(t & 31);
    if (e < E) {
        int s = src[e], d = dst[e];
        atomicAdd(&out[(size_t)d * 32 + f], in[(size_t)s * 32 + f]);
    }
}

// out[i][f] = out[i][f] * norm_d[i] + b3[f]   (F=32)
__global__ __launch_bounds__(256) void k_finalize(float* __restrict__ out,
                                                  const float* __restrict__ norm_d,
                                                  const float* __restrict__ b3, size_t n32) {
    size_t i = (size_t)blockIdx.x * blockDim.x + threadIdx.x;
    if (i < n32) {
        int row = (int)(i >> 5);
        int col = (int)(i & 31);
        out[i] = out[i] * norm_d[row] + b3[col];
    }
}

// ---------------- WMMA GEMM kernels ----------------
// Fragment layout assumption (documented for A/C-D in CDNA5 ISA):
//   A (16x4 f32):  lane l<16 -> row l, K={k0,k0+1} in v0,v1 ; lanes 16-31 -> K={k0+2,k0+3}
//   B (4x16 f32):  mirrored (same per-lane K pair, N = lane&15 striped across lanes)
//   C/D (16x16):   VGPR i, half h -> row i+8h, col lane&15
// A and B use identical per-lane K indices, so the result is invariant to any
// consistent hardware K-pair permutation.

// H1out[N x 128] = relu( (Agg[N x 64] @ W1[64 x 128]) * norm_d[row] + b1[col] )
// 256 threads = 8 waves; each wave owns one 16x16 output tile (8 tiles = 128 cols).
__global__ __launch_bounds__(256) void k_gemm1_wmma(const float* __restrict__ A,
                                                    const float* __restrict__ W,
                                                    const float* __restrict__ bias,
                                                    const float* __restrict__ norm_d,
                                                    float* __restrict__ H, int nrows) {
    int wave = threadIdx.x >> 5;
    int lane = threadIdx.x & 31;
    int half = lane >> 4;     // 0: K pair {0,1}, 1: K pair {2,3}
    int l16  = lane & 15;
    int r0 = blockIdx.x * 16;
    int c0 = wave * 16;
    if (r0 + 16 > nrows) return;   // uniform guard (N is a multiple of 16)

    const float* arow = A + (size_t)(r0 + l16) * 64;
    v8f acc = {};
#pragma unroll
    for (int k0 = 0; k0 < 64; k0 += 4) {
        int ka = k0 + half * 2;
        v2f a;
        a.x = arow[ka];
        a.y = arow[ka + 1];
        v2f b;
        b.x = W[(size_t)ka * 128 + c0 + l16];
        b.y = W[(size_t)(ka + 1) * 128 + c0 + l16];
        acc = __builtin_amdgcn_wmma_f32_16x16x4_f32(false, a, false, b,
                                                    (short)0, acc, false, false);
    }
    int col = c0 + l16;
    float bv = bias[col];
#pragma unroll
    for (int i = 0; i < 8; i++) {
        int row = r0 + i + half * 8;
        float v = acc[i] * norm_d[row] + bv;
        H[(size_t)row * 128 + col] = v > 0.0f ? v : 0.0f;
    }
}

// T3[N x 32] = (H1[N x 128] * norm_s[row]) @ W3[128 x 32]
// 64 threads = 2 waves; each wave owns one 16x16 tile (2 tiles = 32 cols).
__global__ __launch_bounds__(64) void k_gemm2_wmma(const float* __restrict__ A,
                                                   const float* __restrict__ W,
                                                   const float* __restrict__ norm_s,
                                                   float* __restrict__ T, int nrows) {
    int wave = threadIdx.x >> 5;
    int lane = threadIdx.x & 31;
    int half = lane >> 4;
    int l16  = lane & 15;
    int r0 = blockIdx.x * 16;
    int c0 = wave * 16;
    if (r0 + 16 > nrows) return;

    const float* arow = A + (size_t)(r0 + l16) * 128;
    float s = norm_s[r0 + l16];     // prescale folded into the A fragment
    v8f acc = {};
#pragma unroll
    for (int k0 = 0; k0 < 128; k0 += 4) {
        int ka = k0 + half * 2;
        v2f a;
        a.x = arow[ka] * s;
        a.y = arow[ka + 1] * s;
        v2f b;
        b.x = W[(size_t)ka * 32 + c0 + l16];
        b.y = W[(size_t)(ka + 1) * 32 + c0 + l16];
        acc = __builtin_amdgcn_wmma_f32_16x16x4_f32(false, a, false, b,
                                                    (short)0, acc, false, false);
    }
    int col = c0 + l16;
#pragma unroll
    for (int i = 0; i < 8; i++) {
        int row = r0 + i + half * 8;
        T[(size_t)row * 32 + col] = acc[i];
    }
}

// ---------------- launcher ----------------

static inline int blocks_for(size_t n, int bs) { return (int)((n + bs - 1) / bs); }

extern "C" void kernel_launch(void* const* d_in, const int* in_sizes, int n_in,
                              void* d_out, int out_size, void* d_ws, size_t ws_size,
                              hipStream_t stream) {
    const float* x  = (const float*)d_in[0];   // [N,64]
    const float* W1 = (const float*)d_in[1];   // [64,128]
    const float* b1 = (const float*)d_in[2];   // [128]
    const float* W3 = (const float*)d_in[3];   // [128,32]
    const float* b3 = (const float*)d_in[4];   // [32]
    const int* src  = (const int*)d_in[5];     // [E]
    const int* dst  = (const int*)d_in[6];     // [E]
    float* out = (float*)d_out;                // [N,32]

    const int N = in_sizes[0] / 64;
    const int E = in_sizes[5];
    const size_t Ns = (size_t)N;

    // workspace layout (floats): norm_s[N] norm_d[N] hA[64N] hB[64N] h1[128N]
    float* ws     = (float*)d_ws;
    float* norm_s = ws;
    float* norm_d = ws + Ns;
    float* hA     = ws + 2 * Ns;
    float* hB     = hA + 64 * Ns;
    float* h1     = hB + 64 * Ns;
    float* t3     = hA;            // hA is dead once the agg1 scatter finishes

    const int BS = 256;

    // 1) degrees -> norms  (deg accumulated in the norm buffers, rsqrt in place)
    k_zero<<<blocks_for(2 * Ns, BS), BS, 0, stream>>>(norm_s, 2 * Ns);
    k_degree<<<blocks_for((size_t)E, BS), BS, 0, stream>>>(src, dst, norm_s, norm_d, E);
    k_rsqrt<<<blocks_for(2 * Ns, BS), BS, 0, stream>>>(norm_s, 2 * Ns);

    // 2) three residual message layers: nxt = cur + seg_sum(cur[src], dst)
    const size_t nf64 = Ns * 64;
    const size_t ef64 = (size_t)E * 64;
    // layer 1: x -> hA
    k_copy<<<blocks_for(nf64, BS), BS, 0, stream>>>(hA, x, nf64);
    k_scatter64<<<blocks_for(ef64, BS), BS, 0, stream>>>(x, src, dst, hA, E);
    // layer 2: hA -> hB
    k_copy<<<blocks_for(nf64, BS), BS, 0, stream>>>(hB, hA, nf64);
    k_scatter64<<<blocks_for(ef64, BS), BS, 0, stream>>>(hA, src, dst, hB, E);
    // layer 3: hB -> hA
    k_copy<<<blocks_for(nf64, BS), BS, 0, stream>>>(hA, hB, nf64);
    k_scatter64<<<blocks_for(ef64, BS), BS, 0, stream>>>(hB, src, dst, hA, E);

    // 3) GraphConv1: agg = seg_sum((hA*norm_s)[src], dst) -> hB (aggregate-first path)
    k_zero<<<blocks_for(nf64, BS), BS, 0, stream>>>(hB, nf64);
    k_scatter64_scaled<<<blocks_for(ef64, BS), BS, 0, stream>>>(hA, norm_s, src, dst, hB, E);
    //    h1 = relu(agg @ W1 * norm_d + b1)   [WMMA, fused epilogue]
    k_gemm1_wmma<<<N / 16, 256, 0, stream>>>(hB, W1, b1, norm_d, h1, N);

    // 4) GraphConv2 (matmul-first path): t3 = (h1*norm_s) @ W3   [WMMA, fused prescale]
    k_gemm2_wmma<<<N / 16, 64, 0, stream>>>(h1, W3, norm_s, t3, N);

    //    out = seg_sum(t3[src], dst) * norm_d + b3
    const size_t nf32 = Ns * 32;
    const size_t ef32 = (size_t)E * 32;
    k_zero<<<blocks_for(nf32, BS), BS, 0, stream>>>(out, nf32);
    k_scatter32<<<blocks_for(ef32, BS), BS, 0, stream>>>(t3, src, dst, out, E);
    k_finalize<<<blocks_for(nf32, BS), BS, 0, stream>>>(out, norm_d, b3, nf32);
}